// SpiralAutoencoder_62079457296873
// MI455X (gfx1250) — compile-verified
//
#include <hip/hip_runtime.h>
#include <math.h>

typedef __attribute__((ext_vector_type(2))) float v2f;
typedef __attribute__((ext_vector_type(8))) float v8f;
typedef __attribute__((ext_vector_type(4))) int   v4i;

#define TF 64      // audio frames
#define HD 256     // LSTM hidden

// Spiral GEMM K-chunking: KC divides 288/576/1152; KCP keeps rows 16B aligned.
#define KC  288
#define KCP (KC + 4)

#if __has_builtin(__builtin_amdgcn_global_load_async_to_lds_b128) && \
    __has_builtin(__builtin_amdgcn_s_wait_asynccnt)
#define USE_ASYNC_LDS 1
typedef __attribute__((address_space(1))) v4i gv4i;   // global 16B packet
typedef __attribute__((address_space(3))) v4i lv4i;   // LDS 16B packet
#else
#define USE_ASYNC_LDS 0
#endif

__device__ __forceinline__ float sigm_(float x) { return 1.0f / (1.0f + expf(-x)); }

// ---------------------------------------------------------------------------
// Generic C[M,N] = A[M,K] @ W[N,K]^T + bias[N], f32 WMMA 16x16x4.
// Grid: (N/16, M/16), block: 32 (one wave per 16x16 tile). K % 4 == 0.
// ---------------------------------------------------------------------------
__global__ void gemm_nt_wmma(const float* __restrict__ A, const float* __restrict__ W,
                             const float* __restrict__ bias, float* __restrict__ C,
                             int M, int N, int K)
{
    const int lane = threadIdx.x & 31;
    const int mt = blockIdx.y, nt = blockIdx.x;
    const int lr = lane & 15;   // row-in-tile (A) / col-in-tile (B)
    const int kh = lane >> 4;   // which K-half this lane supplies
    const float* __restrict__ arow = A + (size_t)(mt * 16 + lr) * K;
    const float* __restrict__ brow = W + (size_t)(nt * 16 + lr) * K;
    v8f acc = {};
    #pragma unroll 4
    for (int kb = 0; kb < K; kb += 4) {
        const int k0 = kb + 2 * kh;
        v2f a; a.x = arow[k0]; a.y = arow[k0 + 1];
        v2f b; b.x = brow[k0]; b.y = brow[k0 + 1];
        acc = __builtin_amdgcn_wmma_f32_16x16x4_f32(false, a, false, b,
                                                    (short)0, acc, false, false);
    }
    const int cn = nt * 16 + lr;
    const float bv = bias ? bias[cn] : 0.0f;
    #pragma unroll
    for (int i = 0; i < 8; ++i) {
        const int cm = mt * 16 + i + 8 * kh;
        C[(size_t)cm * N + cn] = acc[i] + bv;
    }
}

// ---------------------------------------------------------------------------
// Transpose nmat matrices [1024,256] -> [256,1024] (coalesced recurrent loads)
// Grid: (1024, nmat), block 256.
// ---------------------------------------------------------------------------
__global__ void transpose_whh(const float* __restrict__ in, float* __restrict__ out, int nmat)
{
    const int m = blockIdx.y;
    const int i = blockIdx.x * blockDim.x + threadIdx.x;   // index into out
    if (m < nmat && i < 1024 * 256) {
        const int k = i >> 10;         // 0..255
        const int r = i & 1023;        // 0..1023
        out[(size_t)m * 262144 + i] = in[(size_t)m * 262144 + r * 256 + k];
    }
}

// ---------------------------------------------------------------------------
// Serial LSTM recurrence, one direction per workgroup (grid=2, block=1024).
// G:    [64, 2048] precomputed Wih@x + b (cols 0..1023 = dir0, 1024.. = dir1)
// WhhT: [2][256][1024] transposed recurrent weights
// zout: [64, 512]  (cols 0..255 fwd, 256..511 bwd)  -- torch gate order ifgo
// ---------------------------------------------------------------------------
__global__ void __launch_bounds__(1024) lstm_recurrent(
    const float* __restrict__ G, const float* __restrict__ WhhT,
    float* __restrict__ zout)
{
    const int d = blockIdx.x;
    const float* __restrict__ Wt = WhhT + (size_t)d * 256 * 1024;
    __shared__ float h[HD], c[HD], gp[4 * HD];
    const int tid = threadIdx.x;
    if (tid < HD) { h[tid] = 0.f; c[tid] = 0.f; }
    __syncthreads();
    for (int s = 0; s < TF; ++s) {
        const int t = d ? (TF - 1 - s) : s;
        float acc = G[t * 2048 + d * 1024 + tid];
        #pragma unroll 8
        for (int k = 0; k < HD; ++k)
            acc += Wt[k * 1024 + tid] * h[k];
        gp[tid] = acc;
        __syncthreads();
        if (tid < HD) {
            const float ig = sigm_(gp[tid]);
            const float fg = sigm_(gp[HD + tid]);
            const float gg = tanhf(gp[2 * HD + tid]);
            const float og = sigm_(gp[3 * HD + tid]);
            const float cn = fg * c[tid] + ig * gg;
            c[tid] = cn;
            const float hn = og * tanhf(cn);
            h[tid] = hn;
            zout[t * 512 + d * HD + tid] = hn;
        }
        __syncthreads();
    }
}

// ---------------------------------------------------------------------------
// Sparse up-transform: out[t,r,:] = sum_{k<3} vals[3r+k] * x[t, cols[3r+k], :]
// Grid: (Vout, T), block: C threads.
// ---------------------------------------------------------------------------
__global__ void pool_sparse(const float* __restrict__ x, const int* __restrict__ cols,
                            const float* __restrict__ vals, float* __restrict__ out,
                            int Vin, int C)
{
    const int r = blockIdx.x, t = blockIdx.y, ct = threadIdx.x;
    const float* __restrict__ xt = x + (size_t)t * Vin * C;
    float a = 0.f;
    #pragma unroll
    for (int k = 0; k < 3; ++k)
        a += vals[3 * r + k] * xt[(size_t)cols[3 * r + k] * C + ct];
    out[((size_t)t * gridDim.x + r) * C + ct] = a;
}

// ---------------------------------------------------------------------------
// Spiral conv as gather-GEMM with K chunked (KC) and double-buffered LDS
// staging (async global->LDS B128 when available).
//   out[t,v,o] = act( sum_{l,c} X[t,idx[v,l],c] * W[o, l*C+c] + bias[o] )
// Grid: T*V/16 (exact); block 128 (4 waves); dyn LDS = 2*16*KCP*4 bytes.
// ---------------------------------------------------------------------------
__global__ void spiral_wmma(const float* __restrict__ X, const int* __restrict__ idx,
                            const float* __restrict__ W, const float* __restrict__ bias,
                            const float* __restrict__ actor, float* __restrict__ out,
                            int V, int C, int OC, int do_elu)
{
    extern __shared__ float atile[];              // [2][16][KCP]
    const int K   = 9 * C;
    const int nchunks = K / KC;                   // exact: K in {288,576,1152}
    const int tid = threadIdx.x;
    const int m0  = blockIdx.x * 16;              // T*V multiple of 16 -> no edge

    // ---- chunk staging: gather 16 spiral rows x KC into LDS buffer b ----
    auto stage = [&](int ci, int b) {
        float* __restrict__ dst = atile + b * 16 * KCP;
        const int ngrp = 16 * (KC / 4);           // 16B packets
        for (int g = tid; g < ngrp; g += blockDim.x) {
            const int r  = g / (KC / 4);
            const int k  = (g - r * (KC / 4)) * 4;       // k within chunk
            const int kk = ci * KC + k;                  // absolute k
            const int m  = m0 + r;
            const int t  = m / V, vv = m - t * V;
            const int l  = kk / C, cc = kk - l * C;
            const int src = idx[vv * 9 + l];
            const float* gp = X + ((size_t)t * V + src) * C + cc;
            float* lp = dst + r * KCP + k;
#if USE_ASYNC_LDS
            __builtin_amdgcn_global_load_async_to_lds_b128((gv4i*)gp, (lv4i*)lp, 0, 0);
#else
            *(float4*)lp = *(const float4*)gp;
#endif
        }
    };

    const int lane = tid & 31;
    const int wave = tid >> 5;
    const int nwaves = blockDim.x >> 5;
    const int lr = lane & 15;
    const int kh = lane >> 4;
    const int ntiles = (OC + 15) >> 4;

    // per-wave static N-tile assignment (<= 2 tiles/wave for OC<=128)
    v8f acc[2]; acc[0] = (v8f){}; acc[1] = (v8f){};
    int   ntid[2];  float nval[2];  const float* wr[2];
    int myNT = 0;
    for (int j = 0; j < 2; ++j) {
        const int nt = wave + j * nwaves;
        if (nt < ntiles) {
            const int n = nt * 16 + lr;
            ntid[myNT] = nt;
            nval[myNT] = (n < OC) ? 1.0f : 0.0f;
            wr[myNT]   = W + (size_t)((n < OC) ? n : 0) * K;
            ++myNT;
        }
    }

    stage(0, 0);
#if USE_ASYNC_LDS
    __builtin_amdgcn_s_wait_asynccnt(0);
#endif
    __syncthreads();

    for (int ci = 0; ci < nchunks; ++ci) {
        if (ci + 1 < nchunks) {
            stage(ci + 1, (ci + 1) & 1);          // fire next chunk (async)
            for (int j = 0; j < myNT; ++j)        // prefetch next weight slice
                __builtin_prefetch(wr[j] + (ci + 1) * KC, 0, 1);
        }
        const float* __restrict__ ab = atile + (ci & 1) * 16 * KCP;
        for (int j = 0; j < myNT; ++j) {
            v8f a8 = acc[j];
            const float* __restrict__ wj = wr[j] + ci * KC;
            const float  nv = nval[j];
            #pragma unroll 2
            for (int kb = 0; kb < KC; kb += 4) {
                const int k0 = kb + 2 * kh;
                v2f a; a.x = ab[lr * KCP + k0]; a.y = ab[lr * KCP + k0 + 1];
                v2f b; b.x = wj[k0] * nv;       b.y = wj[k0 + 1] * nv;
                a8 = __builtin_amdgcn_wmma_f32_16x16x4_f32(false, a, false, b,
                                                           (short)0, a8, false, false);
            }
            acc[j] = a8;
        }
#if USE_ASYNC_LDS
        __builtin_amdgcn_s_wait_asynccnt(0);
#endif
        __syncthreads();                          // readers done + next chunk visible
    }

    const int TV = TF * V;
    for (int j = 0; j < myNT; ++j) {
        const int n = ntid[j] * 16 + lr;
        if (n >= OC) continue;
        const float bv = bias[n];
        #pragma unroll
        for (int i = 0; i < 8; ++i) {
            const int m = m0 + i + 8 * kh;
            if (m < TV) {
                float val = acc[j][i] + bv;
                if (do_elu) val = (val > 0.f) ? val : expm1f(val);
                if (actor)  val += actor[(m % V) * 3 + n];
                out[(size_t)m * OC + n] = val;
            }
        }
    }
}

// ---------------------------------------------------------------------------
extern "C" void kernel_launch(void* const* d_in, const int* in_sizes, int n_in,
                              void* d_out, int out_size, void* d_ws, size_t ws_size,
                              hipStream_t stream)
{
    (void)in_sizes; (void)n_in; (void)out_size; (void)ws_size;

    const float* audio = (const float*)d_in[0];
    const float* actor = (const float*)d_in[1];
    const float* Wih0  = (const float*)d_in[2];   // [2048, 768]
    const float* Whh0  = (const float*)d_in[3];   // [2, 1024, 256]
    const float* b0    = (const float*)d_in[4];   // [2048]
    const float* Wih   = (const float*)d_in[5];   // [4][2048, 512]
    const float* Whh   = (const float*)d_in[6];   // [4][2, 1024, 256]
    const float* bl    = (const float*)d_in[7];   // [4][2048]
    const float* fcW   = (const float*)d_in[8];   // [2560, 512]
    const float* fcb   = (const float*)d_in[9];
    const float* c3W = (const float*)d_in[10]; const float* c3b = (const float*)d_in[11];
    const float* c2W = (const float*)d_in[12]; const float* c2b = (const float*)d_in[13];
    const float* c1W = (const float*)d_in[14]; const float* c1b = (const float*)d_in[15];
    const float* c0W = (const float*)d_in[16]; const float* c0b = (const float*)d_in[17];
    const float* hW  = (const float*)d_in[18]; const float* hb  = (const float*)d_in[19];
    const int* idx0 = (const int*)d_in[20]; const int* idx1 = (const int*)d_in[21];
    const int* idx2 = (const int*)d_in[22]; const int* idx3 = (const int*)d_in[23];
    const int*   cols0 = (const int*)d_in[25]; const float* vals0 = (const float*)d_in[26];
    const int*   cols1 = (const int*)d_in[28]; const float* vals1 = (const float*)d_in[29];
    const int*   cols2 = (const int*)d_in[31]; const float* vals2 = (const float*)d_in[32];
    const int*   cols3 = (const int*)d_in[34]; const float* vals3 = (const float*)d_in[35];

    // workspace carve-up (all f32)
    float* WhhT = (float*)d_ws;                  // 10 * 262144
    float* Gpre = WhhT + 10 * 262144;            // 64 * 2048
    float* zA   = Gpre + 64 * 2048;              // 64 * 512
    float* zB   = zA   + 64 * 512;               // 64 * 512
    float* xfc  = zB   + 64 * 512;               // 64 * 2560
    float* bufP = xfc  + 64 * 2560;              // 64 * 5023 * 64 (max pooled)
    float* bufO = bufP + (size_t)64 * 5023 * 64; // 64 * 5023 * 32 (max conv out)

    const int spiralLds = 2 * 16 * KCP * 4;      // 37,376 B double-buffered

    // Pre-transpose recurrent weights for coalesced streaming.
    transpose_whh<<<dim3(1024, 2), 256, 0, stream>>>(Whh0, WhhT, 2);
    transpose_whh<<<dim3(1024, 8), 256, 0, stream>>>(Whh, WhhT + 2 * 262144, 8);

    // ---- bi-LSTM stack ----
    gemm_nt_wmma<<<dim3(2048 / 16, 64 / 16), 32, 0, stream>>>(audio, Wih0, b0, Gpre, 64, 2048, 768);
    lstm_recurrent<<<2, 1024, 0, stream>>>(Gpre, WhhT, zA);
    float* zin = zA; float* zo = zB;
    for (int l = 0; l < 4; ++l) {
        gemm_nt_wmma<<<dim3(128, 4), 32, 0, stream>>>(zin, Wih + (size_t)l * 2048 * 512,
                                                      bl + l * 2048, Gpre, 64, 2048, 512);
        lstm_recurrent<<<2, 1024, 0, stream>>>(Gpre, WhhT + (size_t)(2 + 2 * l) * 262144, zo);
        float* tmp = zin; zin = zo; zo = tmp;
    }

    // ---- decoder entry fc: [64,512] -> [64,2560] = [64, 20, 128] ----
    gemm_nt_wmma<<<dim3(2560 / 16, 4), 32, 0, stream>>>(zin, fcW, fcb, xfc, 64, 2560, 512);

    // ---- level 3: pool 20->79 (C=128), spiral OC=128 ----
    pool_sparse<<<dim3(79, TF), 128, 0, stream>>>(xfc, cols3, vals3, bufP, 20, 128);
    spiral_wmma<<<dim3(TF * 79 / 16), 128, spiralLds, stream>>>(
        bufP, idx3, c3W, c3b, nullptr, bufO, 79, 128, 128, 1);

    // ---- level 2: pool 79->314 (C=128), spiral OC=64 ----
    pool_sparse<<<dim3(314, TF), 128, 0, stream>>>(bufO, cols2, vals2, bufP, 79, 128);
    spiral_wmma<<<dim3(TF * 314 / 16), 128, spiralLds, stream>>>(
        bufP, idx2, c2W, c2b, nullptr, bufO, 314, 128, 64, 1);

    // ---- level 1: pool 314->1256 (C=64), spiral OC=64 ----
    pool_sparse<<<dim3(1256, TF), 64, 0, stream>>>(bufO, cols1, vals1, bufP, 314, 64);
    spiral_wmma<<<dim3(TF * 1256 / 16), 128, spiralLds, stream>>>(
        bufP, idx1, c1W, c1b, nullptr, bufO, 1256, 64, 64, 1);

    // ---- level 0: pool 1256->5023 (C=64), spiral OC=32 ----
    pool_sparse<<<dim3(5023, TF), 64, 0, stream>>>(bufO, cols0, vals0, bufP, 1256, 64);
    spiral_wmma<<<dim3(TF * 5023 / 16), 128, spiralLds, stream>>>(
        bufP, idx0, c0W, c0b, nullptr, bufO, 5023, 64, 32, 1);

    // ---- head: spiral OC=3 (no ELU) + actor offset -> d_out ----
    spiral_wmma<<<dim3(TF * 5023 / 16), 128, spiralLds, stream>>>(
        bufO, idx0, hW, hb, actor, (float*)d_out, 5023, 32, 3, 0);
}